// PolyformerInverseEmbedding_5351529250951
// MI455X (gfx1250) — compile-verified
//
#include <hip/hip_runtime.h>
#include <hip/hip_bf16.h>

typedef float v2f __attribute__((ext_vector_type(2)));
typedef float v8f __attribute__((ext_vector_type(8)));

#define M_DIM 64
#define MM (M_DIM * M_DIM)

// ---------------------------------------------------------------------------
// Kernel 1: Chebyshev basis table Tb[i][k][j] = T_k(t[i,j]),
// t[i,j] = grid[i + j*64].  64 blocks (i) x 64 threads (j).  1 MB in d_ws,
// L2-resident for the GEMM kernel.
// ---------------------------------------------------------------------------
__global__ void cheb_basis_kernel(const float* __restrict__ grid,
                                  float* __restrict__ Tb) {
  const int i = blockIdx.x;
  const int j = threadIdx.x;
  const float t = grid[i + j * M_DIM];
  float* base = Tb + i * MM + j;           // stride M_DIM between k's
  float tkm2 = 1.0f;                       // T_0
  float tkm1 = t;                          // T_1
  base[0] = 1.0f;
  base[M_DIM] = t;
  const float t2 = 2.0f * t;
  for (int k = 2; k < M_DIM; ++k) {
    const float tk = t2 * tkm1 - tkm2;     // T_k = 2 t T_{k-1} - T_{k-2}
    base[k * M_DIM] = tk;
    tkm2 = tkm1;
    tkm1 = tk;
  }
}

// ---------------------------------------------------------------------------
// Kernel 2: pred[:, i, :] = x[:, i, :] @ Tb_i  via V_WMMA_F32_16X16X4_F32.
// Register-blocked: one wave32 computes 64 batches x 16 cols (4 accumulators),
// reusing each B fragment across 4 WMMAs.  64 v_wmma per wave, K-loop 64/4.
// Tile id = bt*256 + i*4 + jt with bt over B/64; 8 waves per 256-thread block.
// Output = pre-inversion matrix, written to d_out; kernel 3 overwrites it.
// Fragment layouts per ISA 7.12.2:
//   A (16x4): lanes 0-15 hold K={k0,k0+1}, lanes 16-31 hold K={k0+2,k0+3}
//   B (4x16): VGPR0 = row k0 / k0+2 across lane halves, VGPR1 = k0+1 / k0+3
//   C/D: VGPR r -> row r (lanes 0-15) / r+8 (lanes 16-31), col = lane&15
// ---------------------------------------------------------------------------
__global__ void cheb_eval_wmma_kernel(const float* __restrict__ x,
                                      const float* __restrict__ Tb,
                                      float* __restrict__ pred) {
  const int lane = threadIdx.x & 31;
  const int wave = threadIdx.x >> 5;
  const int tile = blockIdx.x * 8 + wave;

  const int jt = tile & 3;
  const int i  = (tile >> 2) & (M_DIM - 1);
  const int bt = tile >> 8;
  const int b0 = bt << 6;          // 64 batches per wave
  const int j0 = jt << 4;

  const int half = lane >> 4;      // 0: lanes 0-15, 1: lanes 16-31
  const int lrow = lane & 15;      // batch-row (A) / column n (B,D)

  // A: coefficient rows c[b0+16d+lrow, i, :]; 16 batches apart = 16*MM floats
  const float* __restrict__ arow = x + (size_t)(b0 + lrow) * MM + i * M_DIM;
  // B: basis column j0+lrow of Tb_i (stride M_DIM between k's)
  const float* __restrict__ bcol = Tb + i * MM + j0 + lrow;

  v8f acc0 = {0.f,0.f,0.f,0.f,0.f,0.f,0.f,0.f};
  v8f acc1 = acc0, acc2 = acc0, acc3 = acc0;

#pragma unroll
  for (int k0 = 0; k0 < M_DIM; k0 += 4) {
    const int kk = k0 + half * 2;
    v2f b;
    b[0] = bcol[kk * M_DIM];
    b[1] = bcol[(kk + 1) * M_DIM];
    v2f a0, a1, a2, a3;                       // 8B contiguous pairs
    a0[0] = arow[kk];             a0[1] = arow[kk + 1];
    a1[0] = arow[16 * MM + kk];   a1[1] = arow[16 * MM + kk + 1];
    a2[0] = arow[32 * MM + kk];   a2[1] = arow[32 * MM + kk + 1];
    a3[0] = arow[48 * MM + kk];   a3[1] = arow[48 * MM + kk + 1];
    acc0 = __builtin_amdgcn_wmma_f32_16x16x4_f32(false, a0, false, b,
                                                 (short)0, acc0, false, false);
    acc1 = __builtin_amdgcn_wmma_f32_16x16x4_f32(false, a1, false, b,
                                                 (short)0, acc1, false, false);
    acc2 = __builtin_amdgcn_wmma_f32_16x16x4_f32(false, a2, false, b,
                                                 (short)0, acc2, false, false);
    acc3 = __builtin_amdgcn_wmma_f32_16x16x4_f32(false, a3, false, b,
                                                 (short)0, acc3, false, false);
  }

  float* __restrict__ obase =
      pred + (size_t)b0 * MM + i * M_DIM + j0 + lrow;
#pragma unroll
  for (int r = 0; r < 8; ++r) {
    const size_t rowoff = (size_t)(r + half * 8) * MM;
    obase[rowoff]               = acc0[r];
    obase[rowoff + 16 * MM]     = acc1[r];
    obase[rowoff + 32 * MM]     = acc2[r];
    obase[rowoff + 48 * MM]     = acc3[r];
  }
}

// ---------------------------------------------------------------------------
// Kernel 3: in-place Gauss-Jordan inversion with partial pivoting.
// One 256-thread block (8 wave32) per batch; [A | I] in LDS (row pad 65).
// Each thread owns one column c = tid&127 of the augmented matrix and one
// row parity r0 = tid>>7: the only writer of its column -> pivot row value
// cached in a register per step; pivot-row scaling folded into elimination.
// Pivot search: wave-0 shuffle max-reduce.  4 barriers per column step.
// ---------------------------------------------------------------------------
__global__ void invert64_kernel(float* __restrict__ io) {
  __shared__ float A[M_DIM][M_DIM + 1];
  __shared__ float Inv[M_DIM][M_DIM + 1];
  __shared__ float fcol[M_DIM];
  __shared__ float pvs;
  __shared__ int piv;

  const int tid = threadIdx.x;
  const int c  = tid & 127;        // owned column of [A | Inv]
  const int r0 = tid >> 7;         // row parity (0/1)
  const size_t base = (size_t)blockIdx.x * MM;

  for (int idx = tid; idx < MM; idx += 256) {
    const int r = idx >> 6, cc = idx & 63;
    A[r][cc] = io[base + idx];
    Inv[r][cc] = (r == cc) ? 1.0f : 0.0f;
  }
  __syncthreads();

  // pointer to row 0 of the owned column; row stride = 65 floats
  float* const colptr = (c < M_DIM) ? &A[0][c] : &Inv[0][c - M_DIM];

  for (int col = 0; col < M_DIM; ++col) {
    // --- partial pivot search: wave 0, 2 rows/lane + 5-step shuffle reduce ---
    if (tid < 32) {
      float v = (tid >= col) ? fabsf(A[tid][col]) : -1.0f;
      int bi = tid;
      const int r2 = tid + 32;
      const float v2 = (r2 >= col) ? fabsf(A[r2][col]) : -1.0f;
      if (v2 > v) { v = v2; bi = r2; }
#pragma unroll
      for (int off = 16; off > 0; off >>= 1) {
        const float ov = __shfl_xor(v, off, 32);
        const int   oi = __shfl_xor(bi, off, 32);
        if (ov > v) { v = ov; bi = oi; }
      }
      if (tid == 0) piv = bi;
    }
    __syncthreads();

    // --- row swap in [A | Inv] (each thread swaps its own column) ---
    const int p = piv;
    if (p != col && tid < 128) {
      const float t = colptr[col * (M_DIM + 1)];
      colptr[col * (M_DIM + 1)] = colptr[p * (M_DIM + 1)];
      colptr[p * (M_DIM + 1)] = t;
    }
    __syncthreads();

    // --- snapshot pivot column, reciprocal, unscaled pivot-row element ---
    if (tid == 0) pvs = 1.0f / A[col][col];
    if (tid < M_DIM) fcol[tid] = A[tid][col];
    const float pr_u = colptr[col * (M_DIM + 1)];   // pre-barrier: no writer yet
    __syncthreads();

    // --- elimination + pivot-row scaling (owner thread writes row `col`) ---
    const float prow = pr_u * pvs;                  // scaled pivot-row value
    for (int r = r0; r < M_DIM; r += 2) {
      if (r == col) {
        colptr[r * (M_DIM + 1)] = prow;
      } else {
        const float f = fcol[r];
        colptr[r * (M_DIM + 1)] = fmaf(-f, prow, colptr[r * (M_DIM + 1)]);
      }
    }
    __syncthreads();
  }

  for (int idx = tid; idx < MM; idx += 256) {
    io[base + idx] = Inv[idx >> 6][idx & 63];
  }
}

// ---------------------------------------------------------------------------
extern "C" void kernel_launch(void* const* d_in, const int* in_sizes, int n_in,
                              void* d_out, int out_size, void* d_ws, size_t ws_size,
                              hipStream_t stream) {
  const float* x    = (const float*)d_in[0];   // [B, 64, 64] fp32 coefficients
  const float* grid = (const float*)d_in[1];   // [4096] fp32 forecasting grid
  float* out = (float*)d_out;                  // [B, 64, 64] fp32 inverse
  float* Tb  = (float*)d_ws;                   // 64*64*64 fp32 basis (1 MB)

  const int Bsz = in_sizes[0] / MM;            // 8192

  // 1) basis table
  cheb_basis_kernel<<<M_DIM, M_DIM, 0, stream>>>(grid, Tb);

  // 2) Chebyshev eval as 64 GEMMs of [B,64]x[64,64] via fp32 WMMA,
  //    register-blocked 64 batches x 16 cols per wave
  const int tiles = (Bsz / 64) * M_DIM * 4;    // 32768 wave-tiles
  cheb_eval_wmma_kernel<<<tiles / 8, 256, 0, stream>>>(x, Tb, out);

  // 3) per-batch 64x64 Gauss-Jordan inversion, in place on d_out
  invert64_kernel<<<Bsz, 256, 0, stream>>>(out);
}